// TokenPooling_44057774522435
// MI455X (gfx1250) — compile-verified
//
#include <hip/hip_runtime.h>
#include <stdint.h>

#ifndef __has_builtin
#define __has_builtin(x) 0
#endif

#define TPK_B 32
#define TPK_N 4096
#define TPK_D 768
#define TPK_K 1024
#define TPK_RPB 4   // rows per gather block

// Native clang vector types (HIP's float4 is a struct and is rejected by
// __builtin_nontemporal_load/store; these are true vectors).
typedef int   tpk_v4i __attribute__((vector_size(4 * sizeof(int))));
typedef float tpk_v4f __attribute__((ext_vector_type(4)));

typedef const __attribute__((address_space(1))) float* tpk_gfp;   // global float*
typedef __attribute__((address_space(3))) float*       tpk_lfp;   // lds float*
typedef __attribute__((address_space(1))) tpk_v4i*     tpk_gv4;   // global v4i*
typedef __attribute__((address_space(3))) tpk_v4i*     tpk_lv4;   // lds v4i*

// Pack (value, index) into a u64 so that ascending u64 sort yields
// descending float value with ascending-index tie break (== jax.lax.top_k).
__device__ __forceinline__ uint64_t tpk_key(float f, uint32_t idx) {
  uint32_t u = __float_as_uint(f);
  uint32_t ordv = (u & 0x80000000u) ? ~u : (u | 0x80000000u);  // float -> monotone u32
  uint32_t hi = ~ordv;  // ascending sort on hi -> descending float value
  return ((uint64_t)hi << 32) | (uint64_t)idx;
}

// One block per batch: stage significance row into LDS via the async
// global->LDS path (ASYNCcnt tracked), build 64-bit keys, bitonic-sort
// 4096 keys in LDS, emit top-1024 indices in descending-significance order.
__global__ __launch_bounds__(1024) void tpk_sort_kernel(
    const float* __restrict__ sig, uint32_t* __restrict__ idx_out) {
  __shared__ uint64_t keys[TPK_N];   // 32 KB
  __shared__ float    stage[TPK_N];  // 16 KB
  const uint32_t t = threadIdx.x;
  const uint32_t b = blockIdx.x;
  const float* row = sig + (size_t)b * TPK_N;

#if __has_builtin(__builtin_amdgcn_global_load_async_to_lds_b128)
  // 16 bytes per lane, memory -> LDS without touching VGPRs.
  __builtin_amdgcn_global_load_async_to_lds_b128(
      (tpk_gv4)(tpk_gfp)(row + 4u * t),
      (tpk_lv4)(tpk_lfp)(stage + 4u * t),
      0, 0);
#if __has_builtin(__builtin_amdgcn_s_wait_asynccnt)
  __builtin_amdgcn_s_wait_asynccnt(0);
#else
  asm volatile("s_wait_asynccnt 0" ::: "memory");
#endif
#else
  ((tpk_v4f*)stage)[t] = ((const tpk_v4f*)row)[t];
#endif
  __syncthreads();

  tpk_v4f v = ((const tpk_v4f*)stage)[t];
  keys[4u * t + 0u] = tpk_key(v.x, 4u * t + 0u);
  keys[4u * t + 1u] = tpk_key(v.y, 4u * t + 1u);
  keys[4u * t + 2u] = tpk_key(v.z, 4u * t + 2u);
  keys[4u * t + 3u] = tpk_key(v.w, 4u * t + 3u);
  __syncthreads();

  // Bitonic sort, ascending. Keys are unique (index in low bits) -> stable.
  for (uint32_t k = 2; k <= TPK_N; k <<= 1) {
    for (uint32_t j = k >> 1; j > 0; j >>= 1) {
#pragma unroll
      for (uint32_t s = 0; s < 4; ++s) {
        uint32_t i = t + s * 1024u;
        uint32_t p = i ^ j;
        if (p > i) {
          uint64_t a = keys[i];
          uint64_t c = keys[p];
          bool descBlock = (i & k) != 0u;
          if ((a > c) != descBlock) {  // ascending block: swap if a>c
            keys[i] = c;
            keys[p] = a;
          }
        }
      }
      __syncthreads();
    }
  }

  // keys[0..1023] = top-1024, descending significance; low 32 bits = index.
  idx_out[(size_t)b * TPK_K + t] = (uint32_t)keys[t];
}

// 4 rows per block, 192 threads: each thread issues 4 independent b128
// non-temporal loads (one per row) before storing -> 4x bytes in flight per
// wave, which is what sustaining 23.3 TB/s needs. Row-index fetches are
// uniform per block -> scalar loads from the L2-resident index array.
__global__ __launch_bounds__(192) void tpk_gather_kernel(
    const float* __restrict__ x, const uint32_t* __restrict__ idx,
    float* __restrict__ out) {
  const uint32_t t  = threadIdx.x;
  const uint32_t b  = blockIdx.y;
  const uint32_t r0 = blockIdx.x * TPK_RPB;  // 0..1024 (row 0 = CLS)

  const tpk_v4f* __restrict__ xb =
      (const tpk_v4f*)(x + (size_t)b * (TPK_N + 1) * TPK_D);
  tpk_v4f* __restrict__ ob =
      (tpk_v4f*)(out + (size_t)b * (TPK_K + 1) * TPK_D);

  tpk_v4f vals[TPK_RPB];
#pragma unroll
  for (uint32_t q = 0; q < TPK_RPB; ++q) {
    uint32_t r = r0 + q;
    if (r < TPK_K + 1u) {
      uint32_t src_row = (r == 0u) ? 0u : 1u + idx[(size_t)b * TPK_K + (r - 1u)];
      vals[q] = __builtin_nontemporal_load(xb + (size_t)src_row * (TPK_D / 4) + t);
    }
  }
#pragma unroll
  for (uint32_t q = 0; q < TPK_RPB; ++q) {
    uint32_t r = r0 + q;
    if (r < TPK_K + 1u) {
      __builtin_nontemporal_store(vals[q], ob + (size_t)r * (TPK_D / 4) + t);
    }
  }
}

extern "C" void kernel_launch(void* const* d_in, const int* in_sizes, int n_in,
                              void* d_out, int out_size, void* d_ws, size_t ws_size,
                              hipStream_t stream) {
  (void)in_sizes; (void)n_in; (void)out_size; (void)ws_size;
  const float* x   = (const float*)d_in[0];   // (32, 4097, 768) f32
  const float* sig = (const float*)d_in[1];   // (32, 4096) f32
  // d_in[2] = keep_tokens (1024, static in this problem)
  uint32_t* idxws = (uint32_t*)d_ws;          // 32*1024 u32 = 128 KB scratch
  float* out = (float*)d_out;                 // (32, 1025, 768) f32

  tpk_sort_kernel<<<dim3(TPK_B), dim3(1024), 0, stream>>>(sig, idxws);
  tpk_gather_kernel<<<dim3((TPK_K + 1 + TPK_RPB - 1) / TPK_RPB, TPK_B),
                      dim3(192), 0, stream>>>(x, idxws, out);
}